// ShapeFormer_34299608826089
// MI455X (gfx1250) — compile-verified
//
#include <hip/hip_runtime.h>
#include <math.h>
#include <stdint.h>

// Problem constants from the reference
#define B_    64
#define D_    8
#define L_    2048
#define S_    128
#define K_    64
#define E_    64
#define WPAD_ 100
#define R_    (K_ + 2 * WPAD_)   // 264
#define NW_   (R_ - K_ + 1)      // 201
#define INV_NORM 0.001f          // 1/NORM
#define MAX_CI_  3.0f

typedef float v2f __attribute__((ext_vector_type(2)));
typedef float v8f __attribute__((ext_vector_type(8)));

// Low 32 bits of a flat pointer to __shared__ = LDS byte offset
// (addrspacecast LDS->flat puts the LDS offset in the low DWORD).
__device__ __forceinline__ uint32_t lds_off(const void* p) {
    return (uint32_t)(uintptr_t)p;
}

// CDNA5 async copy global -> LDS, 16 bytes per lane, tracked by ASYNCcnt.
__device__ __forceinline__ void async_load_b128(uint32_t lds_byte_addr,
                                                uint32_t global_byte_off,
                                                unsigned long long sbase) {
    asm volatile("global_load_async_to_lds_b128 %0, %1, %2 offset:0"
                 :
                 : "v"(lds_byte_addr), "v"(global_byte_off), "s"(sbase)
                 : "memory");
}

__device__ __forceinline__ void wait_asynccnt0() {
    asm volatile("s_wait_asynccnt 0x0" ::: "memory");
}

// ---------------------------------------------------------------------------
// Kernel 1: one wave (32 lanes) per (b,s) pair.
// Extract region, compute complexity-invariant distance per window, argmin,
// write the winning 64-float window to scratch "best" in [S][B][K] layout
// (so kernel 2's A panel for a given s is one contiguous 16 KB block).
// ---------------------------------------------------------------------------
__global__ __launch_bounds__(32)
void sf_dist_kernel(const float* __restrict__ x,
                    const float* __restrict__ shapelets,
                    const int*   __restrict__ dims,
                    const int*   __restrict__ start_pos,
                    float*       __restrict__ best)
{
    __shared__ float reg[R_];
    __shared__ float cs[R_];
    __shared__ float sh[K_];

    const int tid = threadIdx.x;           // 0..31
    const int blk = blockIdx.x;            // 0..B*S-1
    const int b   = blk / S_;
    const int s   = blk - b * S_;

    const int dim = dims[s];
    const int sp  = start_pos[s];
    const float* base = x + ((size_t)b * D_ + dim) * L_ + sp;

    // Stage region and shapelet in LDS (region base is only 4B-aligned ->
    // plain b32 vector loads, not b128 async).
    for (int i = tid; i < R_; i += 32) reg[i] = base[i];
    for (int i = tid; i < K_; i += 32) sh[i]  = shapelets[s * K_ + i];
    __syncthreads();

    // Prefix sum of squared first-differences (cs[0]=0, cs[i]=sum d2[0..i-1])
    if (tid == 0) {
        float acc = 0.0f;
        cs[0] = 0.0f;
        for (int i = 1; i < R_; ++i) {
            float d = reg[i] - reg[i - 1];
            acc += d * d;
            cs[i] = acc;
        }
    }
    __syncthreads();

    // Shapelet complexity (redundant per-lane; LDS broadcast reads)
    float sci = 0.0f;
    for (int k2 = 0; k2 < K_ - 1; ++k2) {
        float d = sh[k2 + 1] - sh[k2];
        sci += d * d;
    }
    sci = sqrtf(sci + INV_NORM);

    // Each lane scans windows w = tid, tid+32, ... ; track (min CID, idx)
    float bestv = 3.4e38f;
    int   besti = 0;
    for (int w = tid; w < NW_; w += 32) {
        float ed2 = 0.0f;
        #pragma unroll 8
        for (int k2 = 0; k2 < K_; ++k2) {
            float d = reg[w + k2] - sh[k2];
            ed2 = fmaf(d, d, ed2);
        }
        float wsum = cs[w + K_ - 1] - cs[w];
        float piss = sqrtf(wsum + INV_NORM);
        float cf   = fminf(fmaxf(piss, sci) / fminf(piss, sci), MAX_CI_);
        float cid  = sqrtf(ed2) * cf;
        if (cid < bestv) { bestv = cid; besti = w; }   // strict <: keeps first
    }

    // Wave-wide argmin (tie -> smaller index, matching jnp.argmin)
    for (int off = 16; off > 0; off >>= 1) {
        float ov = __shfl_xor(bestv, off, 32);
        int   oi = __shfl_xor(besti, off, 32);
        if (ov < bestv || (ov == bestv && oi < besti)) { bestv = ov; besti = oi; }
    }

    // All lanes agree on besti; copy winning window, [S][B][K] layout
    float* dst = best + ((size_t)s * B_ + b) * K_;
    dst[tid]      = reg[besti + tid];
    dst[tid + 32] = reg[besti + 32 + tid];
}

// ---------------------------------------------------------------------------
// Kernel 2: one block (16 waves, 512 threads) per shapelet s.
// out[b,s,e] = sum_k best[s,b,k]*l1_w[s,k,e] + l1_b[s,e]
//              - (sum_k shapelet[s,k]*l2_w[s,k,e] + l2_b[s,e])
// A (best[s], 64x64) and B (l1_w[s], 64x64) panels are staged to LDS with
// CDNA5 async b128 copies; GEMM runs on V_WMMA_F32_16X16X4_F32.
// ---------------------------------------------------------------------------
__global__ __launch_bounds__(512)
void sf_gemm_kernel(const float* __restrict__ best,
                    const float* __restrict__ shapelets,
                    const float* __restrict__ l1_w,
                    const float* __restrict__ l1_b,
                    const float* __restrict__ l2_w,
                    const float* __restrict__ l2_b,
                    float*       __restrict__ out)
{
    __shared__ __align__(16) float As[B_ * K_];   // best[s]  : 64x64 = 16 KB
    __shared__ __align__(16) float Bs[K_ * E_];   // l1_w[s]  : 64x64 = 16 KB
    __shared__ float corr[E_];

    const int s   = blockIdx.x;
    const int tid = threadIdx.x;

    // --- Async stage: 1024 b128 transfers per panel, 2 per thread ---------
    {
        const unsigned long long aBase =
            (unsigned long long)(uintptr_t)(best + (size_t)s * B_ * K_);
        const unsigned long long bBase =
            (unsigned long long)(uintptr_t)(l1_w + (size_t)s * K_ * E_);
        const uint32_t aL = lds_off(As);
        const uint32_t bL = lds_off(Bs);
        const uint32_t o0 = (uint32_t)tid * 16u;
        const uint32_t o1 = o0 + 512u * 16u;
        async_load_b128(aL + o0, o0, aBase);
        async_load_b128(aL + o1, o1, aBase);
        async_load_b128(bL + o0, o0, bBase);
        async_load_b128(bL + o1, o1, bBase);
    }

    // Overlap: corr[e] = l1_b[s,e] - (shapelet[s,:] . l2_w[s,:,e] + l2_b[s,e])
    if (tid < E_) {
        float acc = l2_b[s * E_ + tid];
        for (int k2 = 0; k2 < K_; ++k2)
            acc = fmaf(shapelets[s * K_ + k2],
                       l2_w[((size_t)s * K_ + k2) * E_ + tid], acc);
        corr[tid] = l1_b[s * E_ + tid] - acc;
    }

    wait_asynccnt0();     // this thread's async copies have landed in LDS
    __syncthreads();      // everyone's copies + corr visible block-wide

    const int wave = tid >> 5;        // 0..15
    const int lane = tid & 31;
    const int mt   = wave >> 2;       // 0..3  (M tile: rows = batch b)
    const int nt   = wave & 3;        // 0..3  (N tile: cols = e)
    const int ln   = lane & 15;
    const int hi   = (lane >> 4) & 1; // lanes 16..31

    // Seed accumulator: every row of the tile gets corr[n]
    const float ci = corr[nt * 16 + ln];
    v8f c;
    #pragma unroll
    for (int r = 0; r < 8; ++r) c[r] = ci;

    const int m = mt * 16 + ln;       // A row (batch index)
    const int n = nt * 16 + ln;       // B/D column (e index)

    // K=64 in 16 steps of 4.
    // A 16x4 f32 layout: lanes 0-15 hold K=0,1 ; lanes 16-31 hold K=2,3
    // B 4x16 f32 layout: lanes 0-15 hold K=0,1 rows ; lanes 16-31 hold K=2,3
    #pragma unroll
    for (int kk = 0; kk < 16; ++kk) {
        const int k0 = kk * 4 + hi * 2;
        v2f a;  a.x  = As[m * K_ + k0];       a.y  = As[m * K_ + k0 + 1];
        v2f bb; bb.x = Bs[k0 * E_ + n];       bb.y = Bs[(k0 + 1) * E_ + n];
        c = __builtin_amdgcn_wmma_f32_16x16x4_f32(
                /*neg_a=*/false, a, /*neg_b=*/false, bb,
                /*c_mod=*/(short)0, c, /*reuse_a=*/false, /*reuse_b=*/false);
    }

    // D 16x16 f32 layout: VGPR r = row (r + hi*8), column = ln
    #pragma unroll
    for (int r = 0; r < 8; ++r) {
        const int mr = mt * 16 + r + hi * 8;
        out[((size_t)mr * S_ + s) * E_ + n] = c[r];
    }
}

// ---------------------------------------------------------------------------
extern "C" void kernel_launch(void* const* d_in, const int* in_sizes, int n_in,
                              void* d_out, int out_size, void* d_ws, size_t ws_size,
                              hipStream_t stream)
{
    (void)in_sizes; (void)n_in; (void)out_size; (void)ws_size;

    const float* x         = (const float*)d_in[0];
    const float* shapelets = (const float*)d_in[1];
    const float* l1_w      = (const float*)d_in[2];
    const float* l1_b      = (const float*)d_in[3];
    const float* l2_w      = (const float*)d_in[4];
    const float* l2_b      = (const float*)d_in[5];
    const int*   dims      = (const int*)d_in[6];
    const int*   start_pos = (const int*)d_in[7];

    float* out  = (float*)d_out;
    float* best = (float*)d_ws;   // S*B*K floats = 2 MB scratch, fully overwritten

    sf_dist_kernel<<<B_ * S_, 32, 0, stream>>>(x, shapelets, dims, start_pos, best);
    sf_gemm_kernel<<<S_, 512, 0, stream>>>(best, shapelets, l1_w, l1_b,
                                           l2_w, l2_b, out);
}